// BaseEncoder_36180804501645
// MI455X (gfx1250) — compile-verified
//
#include <hip/hip_runtime.h>
#include <stdint.h>

// ---------------------------------------------------------------------------
// BaseEncoder: x[4096,512] f32 -> out[4096,512,32] f32
//   out[...,0]    = (x < 0) ? 1 : 0
//   out[...,1+b]  = bit (30-b) of bitcast<int32>(|x|)
// Pure streaming-store kernel: 8 MB in, 256 MB out -> HBM-write bound
// (~11 us at 23.3 TB/s). Strategy: stage 32 KB output tiles in LDS and let
// the CDNA5 Tensor Data Mover (TENSOR_STORE_FROM_LDS, cpol=NT since the
// stream exceeds the 192 MB L2) push them out; fall back to 128-bit
// non-temporal stores if the TDM builtin is unavailable on this toolchain.
// ---------------------------------------------------------------------------

typedef float    v4f __attribute__((ext_vector_type(4)));
typedef uint32_t v4u __attribute__((ext_vector_type(4)));
typedef int      v4i __attribute__((ext_vector_type(4)));
typedef int      v8i __attribute__((ext_vector_type(8)));

#define TPB               256
#define ITERS_PER_BLOCK   8
#define CHUNKS_PER_BLOCK  (TPB * ITERS_PER_BLOCK)      /* 2048 float4 chunks */
#define ELEMS_PER_BLOCK   (CHUNKS_PER_BLOCK / 8)       /* 256 input floats   */
#define TILE_BYTES        (CHUNKS_PER_BLOCK * 16)      /* 32 KB per block    */
#define TILE_DWORDS       (TILE_BYTES / 4)             /* 8192               */

#if __has_builtin(__builtin_amdgcn_tensor_store_from_lds)
#define HAVE_TDM 1
#else
#define HAVE_TDM 0
#endif

// chunk j (0..7) of one element: output channels c = 4j .. 4j+3
__device__ __forceinline__ v4f encode_chunk(float xv, uint32_t j) {
  const uint32_t u  = __float_as_uint(xv);
  const uint32_t c0 = j << 2;
  const float    sgn = (xv < 0.0f) ? 1.0f : 0.0f;
  v4f r;
  r.x = (c0 == 0u) ? sgn : (float)((u >> (31u - c0)) & 1u);
  r.y = (float)((u >> (30u - c0)) & 1u);
  r.z = (float)((u >> (29u - c0)) & 1u);
  r.w = (float)((u >> (28u - c0)) & 1u);
  return r;
}

#if HAVE_TDM
// Each block: compute 256 input elements -> 32 KB tile in LDS, then one wave
// fires a single TDM store of the whole tile to global memory.
__global__ __launch_bounds__(TPB) void encode_tdm(const float* __restrict__ x,
                                                  float* __restrict__ out) {
  __shared__ v4f tile[CHUNKS_PER_BLOCK];

  const uint32_t tid    = threadIdx.x;
  const uint32_t j      = tid & 7u;        // chunk-within-element, fixed per lane
  const uint32_t laneE  = tid >> 3;        // element offset within iteration
  const uint32_t eBase  = blockIdx.x * (uint32_t)ELEMS_PER_BLOCK + laneE;

#pragma unroll
  for (uint32_t it = 0; it < ITERS_PER_BLOCK; ++it) {
    const float xv = x[eBase + it * 32u];                 // base + imm offsets
    tile[it * TPB + tid] = encode_chunk(xv, j);           // ds_store_b128
  }

  __syncthreads();   // whole tile resident in LDS

  if (tid < 32u) {   // wave-uniform: only wave 0 issues the DMA (TDM ignores EXEC)
    const uint64_t gaddr = (uint64_t)(uintptr_t)out +
                           (uint64_t)blockIdx.x * (uint64_t)TILE_BYTES;
    const uint32_t laddr = (uint32_t)(uintptr_t)&tile[0];     // LDS byte offset

    // ---- D# group 0 (128b): count=1, lds_addr, global_addr[56:0], type=2
    v4u g0;
    g0.x = 1u;                                                // count=1, user mode
    g0.y = laddr;
    g0.z = (uint32_t)gaddr;                                   // global_addr[31:0]
    g0.w = (uint32_t)((gaddr >> 32) & 0x1FFFFFFull) | (2u << 30); // addr[56:32] | type=2

    // ---- D# group 1 (256b): 1-row tile of TILE_DWORDS 4-byte elements
    v8i g1;
    g1[0] = (int)(2u << 16);                                  // wg_mask=0, data_size=4B
    g1[1] = (int)((TILE_DWORDS & 0xFFFFu) << 16);             // tensor_dim0[15:0] @ bit48
    g1[2] = (int)(((TILE_DWORDS >> 16) & 0xFFFFu)             // tensor_dim0[31:16]
                  | (1u << 16));                              // tensor_dim1 = 1
    g1[3] = (int)((TILE_DWORDS & 0xFFFFu) << 16);             // tile_dim0 = 8192
    g1[4] = (int)1;                                           // tile_dim1=1, tile_dim2=0
    g1[5] = (int)TILE_DWORDS;                                 // tensor_dim0_stride[31:0]
    g1[6] = (int)((TILE_DWORDS & 0xFFFFu) << 16);             // dim0_str[47:32]=0 | dim1_str[15:0]
    g1[7] = 0;                                                // tensor_dim1_stride[47:16]

    const v4i gz4 = {0, 0, 0, 0};                             // groups 2/3 unused (2D)
    const v8i gz8 = {0, 0, 0, 0, 0, 0, 0, 0};                 // extra group (clang-23 form)

    // cpol = 1 -> TH_STORE_NT: 256 MB stream, don't pollute the 192 MB L2.
    __builtin_amdgcn_tensor_store_from_lds(g0, g1, gz4, gz4, gz8, 1);
#if __has_builtin(__builtin_amdgcn_s_wait_tensorcnt)
    __builtin_amdgcn_s_wait_tensorcnt(0);                     // s_wait_tensorcnt 0
#endif
  }
}
#endif  // HAVE_TDM

// Direct path: one 16-byte output chunk per thread, perfectly coalesced
// 128-bit non-temporal stores (output stream is larger than the 192 MB L2).
__global__ __launch_bounds__(TPB) void encode_direct(const float* __restrict__ x,
                                                     v4f* __restrict__ out,
                                                     unsigned long long nChunks,
                                                     unsigned long long chunkBase) {
  const unsigned long long idx =
      chunkBase + (unsigned long long)blockIdx.x * TPB + threadIdx.x;
  if (idx >= nChunks) return;
  const uint32_t e = (uint32_t)(idx >> 3);
  const uint32_t j = (uint32_t)idx & 7u;
  const v4f r = encode_chunk(x[e], j);
  __builtin_nontemporal_store(r, &out[idx]);
}

extern "C" void kernel_launch(void* const* d_in, const int* in_sizes, int n_in,
                              void* d_out, int out_size, void* d_ws, size_t ws_size,
                              hipStream_t stream) {
  (void)n_in; (void)out_size; (void)d_ws; (void)ws_size;
  const float* x = (const float*)d_in[0];
  float* out = (float*)d_out;

  const unsigned long long N      = (unsigned long long)in_sizes[0]; // 2,097,152
  const unsigned long long chunks = N * 8ull;                        // 16,777,216

#if HAVE_TDM
  const unsigned long long fullBlocks = chunks / CHUNKS_PER_BLOCK;   // 8192
  if (fullBlocks) {
    encode_tdm<<<dim3((unsigned)fullBlocks), dim3(TPB), 0, stream>>>(x, out);
  }
  const unsigned long long rem = chunks - fullBlocks * CHUNKS_PER_BLOCK;
  if (rem) {
    const unsigned nb = (unsigned)((rem + TPB - 1) / TPB);
    encode_direct<<<dim3(nb), dim3(TPB), 0, stream>>>(
        x, (v4f*)out, chunks, fullBlocks * (unsigned long long)CHUNKS_PER_BLOCK);
  }
#else
  const unsigned nb = (unsigned)((chunks + TPB - 1) / TPB);
  encode_direct<<<dim3(nb), dim3(TPB), 0, stream>>>(x, (v4f*)out, chunks, 0ull);
#endif
}